// SquaredExponential_40372692582703
// MI455X (gfx1250) — compile-verified
//
#include <hip/hip_runtime.h>

typedef __attribute__((ext_vector_type(2))) float v2f;
typedef __attribute__((ext_vector_type(4))) float v4f;
typedef __attribute__((ext_vector_type(8))) float v8f;

#define DDIM   64                 // feature dimension (fixed by reference)
#define MDIM   8192               // output columns (fixed by reference)
#define NT     4                  // 4 N-tiles of 16 cols per wave -> 16x64 patch
#define WPB    8                  // waves per block (256 threads, wave32)
#define COLS_PER_BLOCK (WPB * NT * 16)   // 512
#define KSTEPS (DDIM / 4)         // 16 WMMA k-steps

// ---------- Pass 1: row norms into workspace ----------
__global__ void __launch_bounds__(256)
sqexp_norms_kernel(const float* __restrict__ X1, const float* __restrict__ X2,
                   float* __restrict__ sq1, float* __restrict__ sq2,
                   int N, int M) {
  int t = blockIdx.x * blockDim.x + threadIdx.x;
  const float* src;
  float* dst;
  int row;
  if (t < N)          { src = X1; dst = sq1; row = t; }
  else if (t < N + M) { src = X2; dst = sq2; row = t - N; }
  else return;
  const v4f* p = (const v4f*)(src + (long long)row * DDIM);
  float s = 0.0f;
#pragma unroll
  for (int i = 0; i < DDIM / 4; ++i) {
    v4f v = p[i];
    s += v.x * v.x + v.y * v.y + v.z * v.z + v.w * v.w;
  }
  dst[row] = s;
}

// ---------- Pass 2: WMMA f32 cross-term + fused RBF epilogue ----------
__global__ void __launch_bounds__(256)
sqexp_wmma_kernel(const float* __restrict__ X1, const float* __restrict__ X2,
                  const float* __restrict__ sq1, const float* __restrict__ sq2,
                  const float* __restrict__ lp, const float* __restrict__ sfp,
                  float* __restrict__ out) {
  const int lane = threadIdx.x & 31;
  const int wave = threadIdx.x >> 5;
  const bool hi  = lane >= 16;     // lanes 16..31: K+2/K+3 for A/B, rows M+8 for C/D
  const int l16  = lane & 15;

  const int m_base = blockIdx.x * 16;
  const int n_base = blockIdx.y * COLS_PER_BLOCK + wave * (NT * 16);

  const int koff = hi ? 2 : 0;

  // Single base pointers; all tile/k displacements become immediate offsets.
  const float* aB = X1 + (long long)(m_base + l16) * DDIM + koff;
  const float* bB = X2 + (long long)(n_base + l16) * DDIM + koff;  // +t*16 rows = +t*1024 floats

  // Preload the entire A operand stream for this wave (16 x v2f = 32 VGPRs).
  v2f a[KSTEPS];
#pragma unroll
  for (int i = 0; i < KSTEPS; ++i)
    a[i] = *(const v2f*)(aB + 4 * i);

  // Double-buffered B fragments: loads for step i+1 issue before step i's WMMAs.
  v2f b[2][NT];
#pragma unroll
  for (int t = 0; t < NT; ++t)
    b[0][t] = *(const v2f*)(bB + t * (16 * DDIM));

  v8f acc[NT] = {};

#pragma unroll
  for (int i = 0; i < KSTEPS; ++i) {
    if (i + 1 < KSTEPS) {
#pragma unroll
      for (int t = 0; t < NT; ++t)
        b[(i + 1) & 1][t] = *(const v2f*)(bB + t * (16 * DDIM) + 4 * (i + 1));
    }
#pragma unroll
    for (int t = 0; t < NT; ++t)
      acc[t] = __builtin_amdgcn_wmma_f32_16x16x4_f32(
          /*neg_a=*/false, a[i], /*neg_b=*/false, b[i & 1][t],
          /*c_mod=*/(short)0, acc[t], /*reuse_a=*/false, /*reuse_b=*/false);
  }

  // Epilogue scalars (1-element device arrays from the harness).
  const float l    = lp[0];
  const float sf   = sfp[0];
  const float coef = -0.72134752044448170f / (l * l);  // -0.5*log2(e)/l^2
  const float sf2  = sf * sf;

  // sq1 for the 8 rows this half-wave owns (base + constant offsets).
  const float* s1B = sq1 + m_base + (hi ? 8 : 0);
  float s1[8];
#pragma unroll
  for (int r = 0; r < 8; ++r)
    s1[r] = s1B[r];

  const float* s2B = sq2 + n_base + l16;

  // One store base; all 32 stores use compile-time offsets (r*MDIM + t*16).
  float* ob = out + (long long)(m_base + (hi ? 8 : 0)) * MDIM + (n_base + l16);

#pragma unroll
  for (int t = 0; t < NT; ++t) {
    const float s2 = s2B[t * 16];
#pragma unroll
    for (int r = 0; r < 8; ++r) {
      float dist = s1[r] + s2 - 2.0f * acc[t][r];
      dist = fmaxf(dist, 0.0f);
      // sigma_f^2 * exp(-0.5*dist/l^2) == sf2 * exp2(coef*dist)  -> v_exp_f32
      ob[r * MDIM + t * 16] = sf2 * __builtin_amdgcn_exp2f(coef * dist);
    }
  }
}

extern "C" void kernel_launch(void* const* d_in, const int* in_sizes, int n_in,
                              void* d_out, int out_size, void* d_ws, size_t ws_size,
                              hipStream_t stream) {
  const float* X1 = (const float*)d_in[0];
  const float* X2 = (const float*)d_in[1];
  const float* lp = (const float*)d_in[2];   // scalar l (1 element)
  const float* sf = (const float*)d_in[3];   // scalar sigma_f (1 element)
  float* out = (float*)d_out;

  const int N = in_sizes[0] / DDIM;   // 8192
  const int M = in_sizes[1] / DDIM;   // 8192 (== MDIM)

  float* sq1 = (float*)d_ws;
  float* sq2 = sq1 + N;

  // Pass 1: norms (one thread per row; ~4 MB traffic -> noise)
  {
    int total = N + M;
    int blocks = (total + 255) / 256;
    sqexp_norms_kernel<<<blocks, 256, 0, stream>>>(X1, X2, sq1, sq2, N, M);
  }

  // Pass 2: 16x64 patch per wave32, 8 waves/block -> grid (512, 16)
  {
    dim3 grid(N / 16, M / COLS_PER_BLOCK);
    sqexp_wmma_kernel<<<grid, 256, 0, stream>>>(X1, X2, sq1, sq2, lp, sf, out);
  }
}